// LRU_12635793785319
// MI455X (gfx1250) — compile-verified
//
#include <hip/hip_runtime.h>

// LRU cascade filter, B=16 T=1024 F=32 H=64.
// Per-timestep state update recast as S_new = S @ M^T + x_t (x) c using
// V_WMMA_F32_16X16X4_F32 tiles. One block per filterbank f, 4 waves = 4
// H-tiles of 16 columns. State ping-pongs between two LDS buffers so each
// timestep needs only ONE barrier (write target is never concurrently read).
// t-loop unrolled by 2 => buffer parity is static, no address arithmetic.
//
// M[i,i] = sq, M[i,j] = (sq^2-1)*sq^(i-j-1) for j<i, 0 for j>i.
// c_i = g*sq^i injects x_t as a rank-1 K-chunk (only k=0 populated).
// All 16 K-chunks run unconditionally (upper-triangle B entries are exact
// zeros); two independent accumulator chains halve dependent-WMMA depth.

typedef float v2f __attribute__((ext_vector_type(2)));
typedef float v8f __attribute__((ext_vector_type(8)));

#define B_ 16
#define T_ 1024
#define F_ 32
#define H_ 64
#define SP 66   // padded LDS row stride (floats): conflict-free, 8B-aligned pairs

#if defined(__has_builtin)
#if __has_builtin(__builtin_amdgcn_wmma_f32_16x16x4_f32)
#define HAVE_WMMA_F32 1
#endif
#endif

__global__ __launch_bounds__(128) void lru_wmma_kernel(
    const float* __restrict__ x,      // [B,T,F]
    const float* __restrict__ relax,  // [F]
    const float* __restrict__ h0,     // [F,B,H]
    float* __restrict__ out)          // [B,T,F,H] then [F,B,H]
{
    __shared__ float S0buf[B_ * SP];  // state ping
    __shared__ float S1buf[B_ * SP];  // state pong
    __shared__ float Mpow[H_];        // sq^k

    const int f    = blockIdx.x;
    const int tid  = threadIdx.x;
    const int w    = tid >> 5;        // wave id = H tile
    const int lane = tid & 31;
    const int half = lane >> 4;       // 0: lanes 0-15, 1: lanes 16-31
    const int n    = lane & 15;       // tile column (B/D) == row m (A)

    const float rx    = relax[f];
    const float sq    = sqrtf(rx);
    const float g     = sqrtf(1.0f - rx);
    const float sq2m1 = rx - 1.0f;    // sq*sq - 1 exactly

    // ---- init: state from h0 into ping buffer, sq-power table ----
    for (int idx = tid; idx < B_ * H_; idx += 128) {
        int b = idx >> 6, i = idx & 63;
        S0buf[b * SP + i] = h0[(f * B_ + b) * H_ + i];
    }
    if (tid == 0) {
        float p = 1.0f;
        for (int k = 0; k < H_; ++k) { Mpow[k] = p; p *= sq; }
    }
    __syncthreads();

    const int iN = 16 * w + n;        // this lane's global H column index

    // ---- precompute all 16 B chunks of M^T: B[k,col] = M[iN, jc*4+k] ----
    // layout: VGPR r, lane L -> k = r + 2*half, col = L%16
    v2f Bc[16];
    #pragma unroll
    for (int jc = 0; jc < 16; ++jc) {
        int j0 = jc * 4 + 2 * half;
        int j1 = j0 + 1;
        int e0 = iN - j0 - 1; if (e0 < 0) e0 = 0;    // clamped index
        int e1 = iN - j1 - 1; if (e1 < 0) e1 = 0;
        float m0 = sq2m1 * Mpow[e0];
        float m1 = sq2m1 * Mpow[e1];
        v2f bv;
        bv[0] = (j0 > iN) ? 0.0f : ((j0 == iN) ? sq : m0);
        bv[1] = (j1 > iN) ? 0.0f : ((j1 == iN) ? sq : m1);
        Bc[jc] = bv;
    }
    // rank-1 x-injection chunk: only k=0 row populated with c_i = g*sq^i
    const float cI = g * Mpow[iN];
    v2f Bx; Bx[0] = half ? 0.0f : cI; Bx[1] = 0.0f;

    const size_t xstride_b = (size_t)T_ * F_;

    // one timestep: read Ssrc, write Sdst + global out; single end barrier
    auto step = [&](int t, const float* Ssrc, float* Sdst) {
        // prefetch x ahead (gfx1250 global_prefetch_b8)
        if (t + 16 < T_)
            __builtin_prefetch(&x[(size_t)n * xstride_b + (size_t)(t + 16) * F_ + f], 0, 0);

        // A layout: VGPR r, lane L -> row m = L%16, k = r + 2*half
        v2f Ac[16];
        #pragma unroll
        for (int jc = 0; jc < 16; ++jc) {
            const float2 v = *(const float2*)&Ssrc[n * SP + jc * 4 + 2 * half];
            v2f a; a[0] = v.x; a[1] = v.y;
            Ac[jc] = a;
        }
        // unconditional load (valid address for all lanes), branchless select
        float xl = x[(size_t)n * xstride_b + (size_t)t * F_ + f];
        v2f Ax; Ax[0] = half ? 0.0f : xl; Ax[1] = 0.0f;

#if HAVE_WMMA_F32
        v8f acc0 = {0.f, 0.f, 0.f, 0.f, 0.f, 0.f, 0.f, 0.f};
        v8f acc1 = {0.f, 0.f, 0.f, 0.f, 0.f, 0.f, 0.f, 0.f};
        acc0 = __builtin_amdgcn_wmma_f32_16x16x4_f32(
            false, Ax, false, Bx, (short)0, acc0, false, false);
        #pragma unroll
        for (int jc = 0; jc < 16; jc += 2) {
            acc0 = __builtin_amdgcn_wmma_f32_16x16x4_f32(
                false, Ac[jc], false, Bc[jc], (short)0, acc0, false, false);
            acc1 = __builtin_amdgcn_wmma_f32_16x16x4_f32(
                false, Ac[jc + 1], false, Bc[jc + 1], (short)0, acc1, false, false);
        }
        v8f acc = acc0 + acc1;
#else
        // correct VALU fallback: per-element triangular matvec
        v8f acc;
        #pragma unroll
        for (int r = 0; r < 8; ++r) {
            int m = r + 8 * half;
            float xm = x[(size_t)m * xstride_b + (size_t)t * F_ + f];
            float d = cI * xm;
            for (int j = 0; j <= iN; ++j) {
                float mv = (j == iN) ? sq : sq2m1 * Mpow[iN - j - 1];
                d += Ssrc[m * SP + j] * mv;
            }
            acc[r] = d;
        }
#endif
        // write phase: D/C layout (VGPR r, lane L) -> (b = r+8*half, i = iN)
        // Sdst is not read by anyone this step -> no barrier needed before it.
        #pragma unroll
        for (int r = 0; r < 8; ++r) {
            int b = r + 8 * half;
            Sdst[b * SP + iN] = acc[r];
            out[(((size_t)b * T_ + t) * F_ + f) * H_ + iN] = acc[r];
        }
        __syncthreads();   // order: this step's writes before next step's reads
    };

    for (int t = 0; t < T_; t += 2) {
        step(t,     S0buf, S1buf);
        step(t + 1, S1buf, S0buf);
    }

    // ---- final hidden state [F,B,H]: last step (t=1023, odd) wrote S0buf ----
    float* hf = out + (size_t)B_ * T_ * F_ * H_;
    for (int idx = tid; idx < B_ * H_; idx += 128) {
        int b = idx >> 6, i = idx & 63;
        hf[((size_t)f * B_ + b) * H_ + i] = S0buf[b * SP + i];
    }
}

extern "C" void kernel_launch(void* const* d_in, const int* in_sizes, int n_in,
                              void* d_out, int out_size, void* d_ws, size_t ws_size,
                              hipStream_t stream) {
    const float* x     = (const float*)d_in[0];
    const float* relax = (const float*)d_in[1];
    const float* h0    = (const float*)d_in[2];
    float* out         = (float*)d_out;
    (void)in_sizes; (void)n_in; (void)out_size; (void)d_ws; (void)ws_size;
    lru_wmma_kernel<<<F_, 128, 0, stream>>>(x, relax, h0, out);
}